// CTRNN_26027501814410
// MI455X (gfx1250) — compile-verified
//
#include <hip/hip_runtime.h>

typedef __attribute__((ext_vector_type(2))) float v2f;
typedef __attribute__((ext_vector_type(8))) float v8f;
typedef __attribute__((ext_vector_type(4))) int   v4i;

typedef __attribute__((address_space(1))) v4i* gv4ip;   // global 16B vector ptr
typedef __attribute__((address_space(3))) v4i* lv4ip;   // LDS 16B vector ptr

#define T_STEPS 1024
#define BATCH   64
#define IN_DIM  128
#define HID     512
#define BH      (BATCH * HID)          // 32768
#define ALPHA   0.02f
#define NWG2    16                     // persistent workgroups for the scan
#define LDSTR   516                    // padded LDS row stride (floats): 2064B rows, 16B aligned, bank shift 4

// ---------------------------------------------------------------------------
// Phase 0: initialize hidden state (broadcast h0 over batch) + barrier counter
// ---------------------------------------------------------------------------
__global__ void ctrnn_init(const float* __restrict__ h0,
                           float* __restrict__ hbuf0,
                           int* __restrict__ counter) {
    int idx = blockIdx.x * 256 + threadIdx.x;        // 0..BH-1
    if (idx < BH) hbuf0[idx] = h0[idx & (HID - 1)];  // h = idx % 512
    if (idx == 0) *counter = 0;
}

// ---------------------------------------------------------------------------
// Phase 1: x_proj[t*B+b][h] = sum_i x[t,b,i] * W_in[h,i]
// M = 65536, N = 512, K = 128.  FP32 WMMA 16x16x4.
// Each block: one 16-row M tile; 8 waves x 64 columns (4 N-subtiles per wave).
// Result written straight into d_out (phase 2 consumes + overwrites in place).
// ---------------------------------------------------------------------------
__global__ __launch_bounds__(256) void ctrnn_xproj(const float* __restrict__ x,
                                                   const float* __restrict__ Win,
                                                   float* __restrict__ xproj) {
    const int lane = threadIdx.x & 31;
    const int wave = threadIdx.x >> 5;   // 0..7
    const int lm   = lane & 15;
    const int hs   = lane >> 4;          // 0/1 -> which K pair
    const int m0   = blockIdx.x * 16;    // row tile over (t*B+b)
    const int n0   = wave * 64;

    const v8f vzero = {0.f, 0.f, 0.f, 0.f, 0.f, 0.f, 0.f, 0.f};
    v8f acc[4] = {vzero, vzero, vzero, vzero};

    const float* arow = x + (size_t)(m0 + lm) * IN_DIM + hs * 2;

    for (int k0 = 0; k0 < IN_DIM; k0 += 4) {
        v2f a = *(const v2f*)(arow + k0);
#pragma unroll
        for (int s = 0; s < 4; ++s) {
            v2f b = *(const v2f*)(Win + (size_t)(n0 + s * 16 + lm) * IN_DIM + k0 + hs * 2);
            acc[s] = __builtin_amdgcn_wmma_f32_16x16x4_f32(
                false, a, false, b, (short)0, acc[s], false, false);
        }
    }

#pragma unroll
    for (int s = 0; s < 4; ++s)
#pragma unroll
        for (int r = 0; r < 8; ++r) {
            int row = m0 + hs * 8 + r;
            int col = n0 + s * 16 + lm;
            xproj[(size_t)row * HID + col] = acc[s][r];
        }
}

// ---------------------------------------------------------------------------
// Phase 2: persistent recurrent scan.
// 16 WGs x 256 threads (8 waves).  WG g owns output columns [g*32, g*32+32):
// 4 M-tiles x 2 N-tiles = 8 tiles = 1 per wave.  Per step:
//   async-DMA hidden[64x512] (L2) -> LDS (ASYNCcnt path, no VGPR transit),
//   GEMM (f32 WMMA, K=512, 4 independent acc chains),
//   pre = acc + x_proj ; h = (1-a)*h_old + a*sigmoid(pre) ; grid barrier.
// W_h slice (32 rows, 64 KB) async-staged into LDS once before the t loop.
// ---------------------------------------------------------------------------
__global__ __launch_bounds__(256) void ctrnn_scan(const float* __restrict__ Wh,
                                                  float* out,          // [T*BH] xproj->output, then [BH] final hidden
                                                  float* __restrict__ hbuf,   // 2*BH double buffer
                                                  int* __restrict__ counter) {
    __shared__ float lds_h[64 * LDSTR];   // 132,096 B : full hidden state, padded
    __shared__ float lds_w[32 * LDSTR];   //  66,048 B : this WG's W_h rows

    const int tid  = threadIdx.x;
    const int lane = tid & 31;
    const int wave = tid >> 5;           // 0..7
    const int lm   = lane & 15;
    const int hs   = lane >> 4;
    const int g    = blockIdx.x;         // 0..15
    const int m0   = (wave & 3) * 16;    // M tile (batch rows)
    const int ns   = wave >> 2;          // 0/1
    const int n0   = g * 32 + ns * 16;   // N tile (hidden columns)

    // ---- async-stage W_h rows [g*32, g*32+32) once ----
    {
        const float* src = Wh + (size_t)g * 32 * HID;
#pragma unroll
        for (int j = 0; j < 16; ++j) {
            int idx = tid + j * 256;         // 0..4095 float4s
            int row = idx >> 7;              // 128 float4 per row
            int c4  = idx & 127;
            __builtin_amdgcn_global_load_async_to_lds_b128(
                (gv4ip)(src + (size_t)idx * 4),
                (lv4ip)(&lds_w[row * LDSTR + c4 * 4]), 0, 0);
        }
    }

    const v8f vzero = {0.f, 0.f, 0.f, 0.f, 0.f, 0.f, 0.f, 0.f};
    v8f acc0 = vzero, acc1 = vzero, acc2 = vzero, acc3 = vzero;

    const float* ah = &lds_h[(m0 + lm) * LDSTR + hs * 2];
    const float* bw = &lds_w[(ns * 16 + lm) * LDSTR + hs * 2];

    int cur = 0;
#pragma unroll 1
    for (int t = 0; t < T_STEPS; ++t) {
        // ---- async-stage hidden[cur] -> LDS (CDNA5 DMA path, ASYNCcnt) ----
        const float* hsrc = hbuf + (size_t)cur * BH;
#pragma unroll
        for (int j = 0; j < 32; ++j) {
            int idx = tid + j * 256;         // 0..8191 float4s
            int row = idx >> 7;
            int c4  = idx & 127;
            __builtin_amdgcn_global_load_async_to_lds_b128(
                (gv4ip)(hsrc + (size_t)idx * 4),
                (lv4ip)(&lds_h[row * LDSTR + c4 * 4]), 0, 0);
        }

        // ---- overlap: pull this wave's x_proj tile while the DMA flies ----
        size_t obase = (size_t)t * BH;
        float xp[8];
#pragma unroll
        for (int r = 0; r < 8; ++r)
            xp[r] = out[obase + (size_t)(m0 + hs * 8 + r) * HID + (n0 + lm)];

        asm volatile("s_wait_asynccnt 0x0" ::: "memory");
        __syncthreads();

        // ---- 16x16 tile GEMM over K=512, 4 independent WMMA chains ----
#pragma unroll 2
        for (int k0 = 0; k0 < HID; k0 += 16) {
            v2f a0 = *(const v2f*)(ah + k0);
            v2f b0 = *(const v2f*)(bw + k0);
            acc0 = __builtin_amdgcn_wmma_f32_16x16x4_f32(false, a0, false, b0, (short)0, acc0, false, false);
            v2f a1 = *(const v2f*)(ah + k0 + 4);
            v2f b1 = *(const v2f*)(bw + k0 + 4);
            acc1 = __builtin_amdgcn_wmma_f32_16x16x4_f32(false, a1, false, b1, (short)0, acc1, false, false);
            v2f a2 = *(const v2f*)(ah + k0 + 8);
            v2f b2 = *(const v2f*)(bw + k0 + 8);
            acc2 = __builtin_amdgcn_wmma_f32_16x16x4_f32(false, a2, false, b2, (short)0, acc2, false, false);
            v2f a3 = *(const v2f*)(ah + k0 + 12);
            v2f b3 = *(const v2f*)(bw + k0 + 12);
            acc3 = __builtin_amdgcn_wmma_f32_16x16x4_f32(false, a3, false, b3, (short)0, acc3, false, false);
        }

        // ---- epilogue: sigmoid + leaky blend, write output + next hidden ----
        float* hnext = hbuf + (size_t)(cur ^ 1) * BH;
#pragma unroll
        for (int r = 0; r < 8; ++r) {
            int    brow = m0 + hs * 8 + r;
            int    h    = n0 + lm;
            size_t oidx = obase + (size_t)brow * HID + h;
            float  pre  = acc0[r] + acc1[r] + acc2[r] + acc3[r] + xp[r];
            float  sg   = 1.0f / (1.0f + __expf(-pre));
            float  hold = lds_h[brow * LDSTR + h];
            float  hnew = hold * (1.0f - ALPHA) + sg * ALPHA;
            out[oidx] = hnew;
            hnext[(size_t)brow * HID + h] = hnew;
            if (t == T_STEPS - 1)
                out[(size_t)T_STEPS * BH + (size_t)brow * HID + h] = hnew;
        }
        acc0 = vzero; acc1 = vzero; acc2 = vzero; acc3 = vzero;

        // ---- grid-wide barrier (release/acquire, agent scope) ----
        __threadfence();
        __syncthreads();
        if (tid == 0) {
            __hip_atomic_fetch_add(counter, 1, __ATOMIC_RELEASE, __HIP_MEMORY_SCOPE_AGENT);
            int target = NWG2 * (t + 1);
            while (__hip_atomic_load(counter, __ATOMIC_ACQUIRE, __HIP_MEMORY_SCOPE_AGENT) < target) {
                __builtin_amdgcn_s_sleep(1);
            }
        }
        __syncthreads();
        cur ^= 1;
    }
}

// ---------------------------------------------------------------------------
extern "C" void kernel_launch(void* const* d_in, const int* in_sizes, int n_in,
                              void* d_out, int out_size, void* d_ws, size_t ws_size,
                              hipStream_t stream) {
    (void)in_sizes; (void)n_in; (void)out_size; (void)ws_size;
    const float* x   = (const float*)d_in[0];   // [T,B,I]
    const float* Win = (const float*)d_in[1];   // [H,I]
    const float* Wh  = (const float*)d_in[2];   // [H,H]
    const float* h0  = (const float*)d_in[3];   // [1,H]
    float* out  = (float*)d_out;                // [T*B*H] outputs, then [B*H] final hidden
    float* hbuf = (float*)d_ws;                 // 2*BH floats double buffer
    int* counter = (int*)((char*)d_ws + (size_t)2 * BH * sizeof(float));

    ctrnn_init <<<dim3((BH + 255) / 256), dim3(256), 0, stream>>>(h0, hbuf, counter);
    ctrnn_xproj<<<dim3((T_STEPS * BATCH) / 16), dim3(256), 0, stream>>>(x, Win, out);
    ctrnn_scan <<<dim3(NWG2), dim3(256), 0, stream>>>(Wh, out, hbuf, counter);
}